// DownLayer_62517543960909
// MI455X (gfx1250) — compile-verified
//
#include <hip/hip_runtime.h>
#include <hip/hip_bf16.h>
#include <stdint.h>

#define B_ 8
#define N_ 16384
#define C_ 256
#define S_ 4096
#define LN_EPS 1e-5f
#define TILES_ 4            // 16-token tiles per wave (double-buffered)

typedef __attribute__((ext_vector_type(2))) float v2f;
typedef __attribute__((ext_vector_type(8))) float v8f;

// ---------------------------------------------------------------------------
// Kernel 1: per-token stats. One wave (32 threads) handles TILES_ tiles of
// 16 tokens. Tiles stream through two LDS buffers via async b128 copies,
// pipelined against 64 x V_WMMA_F32_16X16X4_F32 per tile with
// B = [w' | ones | 0...]  =>  D[:,0] = x . w' ,  D[:,1] = sum(x).
// Sum(x^2) rides along in co-executing VALU FMAs.
// ---------------------------------------------------------------------------
struct Smem1 {
  float A[2][16 * 260];  // double-buffered tile, rows padded to 260 floats
  float W[256];          // w' = ln_w * conf_w
  float D[8 * 32];       // WMMA accumulator spill
};

__global__ __launch_bounds__(32) void stats_kernel(
    const float* __restrict__ x, const float* __restrict__ ln_w,
    const float* __restrict__ ln_b, const float* __restrict__ conf_w,
    const float* __restrict__ conf_b, float* __restrict__ conf,
    float* __restrict__ mu, float* __restrict__ rstd)
{
  __shared__ Smem1 sm;   // single shared object -> sm.A at LDS offset 0
  const int lane = threadIdx.x;
  const long long tile0 = (long long)blockIdx.x * TILES_;

  // Build w' in LDS; reduce sum(w') and sum(ln_b*conf_w) across the wave.
  float wpart = 0.f, cpart = 0.f;
  for (int i = 0; i < 8; ++i) {
    int c = lane + i * 32;
    float cw = conf_w[c];
    float wv = ln_w[c] * cw;
    sm.W[c] = wv;
    wpart += wv;
    cpart += ln_b[c] * cw;
  }
  for (int d = 16; d > 0; d >>= 1) {
    wpart += __shfl_xor(wpart, d, 32);
    cpart += __shfl_xor(cpart, d, 32);
  }
  const float cbias = cpart + conf_b[0];

  // A-tile 16x4 layout: lanes 0-15 hold K={0,1}, lanes 16-31 hold K={2,3}.
  const int mIdx  = lane & 15;     // token row == B-matrix column index n
  const int khalf = lane >> 4;
  const float f0 = (mIdx == 0) ? 1.0f : 0.0f;   // branchless B selection
  const float f1 = (mIdx == 1) ? 1.0f : 0.0f;

  // Issue one tile's 32 async b128 copies (contiguous 16KB -> padded rows).
  auto issue = [&](int t, int buf) {
    const char* gb = (const char*)(x + (tile0 + t) * 16 * C_);
    const unsigned lbase = (unsigned)(buf * (16 * 260 * 4));  // sm.A at offset 0
    for (int half = 0; half < 2; ++half) {
      for (int r = 0; r < 16; ++r) {
        unsigned loff = lbase + (unsigned)(r * 1040 + half * 512 + lane * 16);
        unsigned long long g =
            (unsigned long long)(gb + r * 1024 + half * 512 + lane * 16);
        asm volatile("global_load_async_to_lds_b128 %0, %1, off"
                     :: "v"(loff), "v"(g) : "memory");
      }
    }
  };

  issue(0, 0);
  for (int t = 0; t < TILES_; ++t) {
    if (t + 1 < TILES_) {
      issue(t + 1, (t + 1) & 1);
      // async loads complete in order: <=32 outstanding => tile t has landed
      asm volatile("s_wait_asynccnt 0x20" ::: "memory");
    } else {
      asm volatile("s_wait_asynccnt 0x0" ::: "memory");
    }
    __syncthreads();

    const float* Ab = sm.A[t & 1];
    v8f acc = {0.f, 0.f, 0.f, 0.f, 0.f, 0.f, 0.f, 0.f};
    float sq = 0.f;
    for (int kb = 0; kb < C_; kb += 4) {
      v2f a = *(const v2f*)&Ab[mIdx * 260 + kb + 2 * khalf];
      v2f w = *(const v2f*)&sm.W[kb + 2 * khalf];
      v2f bv;
      bv.x = fmaf(w.x, f0, f1);
      bv.y = fmaf(w.y, f0, f1);
      acc = __builtin_amdgcn_wmma_f32_16x16x4_f32(
          false, a, false, bv, (short)0, acc, false, false);
      sq = fmaf(a.x, a.x, sq);
      sq = fmaf(a.y, a.y, sq);
    }
    const float sumsq = sq + __shfl_xor(sq, 16, 32);  // token mIdx, lanes 0-15

    // Spill D: VGPR j, lanes 0-15 -> M=j ; lanes 16-31 -> M=j+8.
    for (int j = 0; j < 8; ++j) sm.D[j * 32 + lane] = acc[j];
    __syncthreads();

    if (lane < 16) {
      const int srcl = (lane < 8) ? 0 : 16;
      const float dotw = sm.D[(lane & 7) * 32 + srcl + 0];
      const float sumx = sm.D[(lane & 7) * 32 + srcl + 1];
      const float m    = sumx * (1.0f / C_);
      const float var  = sumsq * (1.0f / C_) - m * m;
      const float r    = rsqrtf(var + LN_EPS);
      const float cf   = r * (dotw - m * wpart) + cbias;
      const long long tok = (tile0 + t) * 16 + lane;
      conf[tok] = cf; mu[tok] = m; rstd[tok] = r;
    }
    __syncthreads();
  }
}

// ---------------------------------------------------------------------------
// Kernel 2: per-batch softmax stats (max and sum of exp over N tokens).
// ---------------------------------------------------------------------------
__global__ __launch_bounds__(256) void softmax_stats_kernel(
    const float* __restrict__ conf, float* __restrict__ bmax,
    float* __restrict__ bsum)
{
  __shared__ float red[256];
  const int b = blockIdx.x, tid = threadIdx.x;
  const float* c = conf + (long long)b * N_;
  float lm = -3.4e38f;
  for (int i = tid; i < N_; i += 256) lm = fmaxf(lm, c[i]);
  red[tid] = lm; __syncthreads();
  for (int s = 128; s > 0; s >>= 1) {
    if (tid < s) red[tid] = fmaxf(red[tid], red[tid + s]);
    __syncthreads();
  }
  const float bm = red[0]; __syncthreads();
  float ls = 0.f;
  for (int i = tid; i < N_; i += 256) ls += __expf(c[i] - bm);
  red[tid] = ls; __syncthreads();
  for (int s = 128; s > 0; s >>= 1) {
    if (tid < s) red[tid] += red[tid + s];
    __syncthreads();
  }
  if (tid == 0) { bmax[b] = bm; bsum[b] = red[0]; }
}

// ---------------------------------------------------------------------------
// Kernel 3: per-batch exact top-k via full bitonic sort of 16384 packed keys
// in 128KB dynamic LDS.  key = (~ord(conf) << 32) | index, sorted ascending
// == conf descending with ties broken by ascending index (jax.lax.top_k).
// ---------------------------------------------------------------------------
__global__ __launch_bounds__(1024) void topk_sort_kernel(
    const float* __restrict__ conf, unsigned* __restrict__ topidx)
{
  extern __shared__ unsigned long long key[];
  const int b = blockIdx.x, tid = threadIdx.x;
  const float* c = conf + (long long)b * N_;
  for (int i = tid; i < N_; i += 1024) {
    unsigned u = __float_as_uint(c[i]);
    u = (u & 0x80000000u) ? ~u : (u | 0x80000000u);   // order-preserving map
    key[i] = ((unsigned long long)(~u) << 32) | (unsigned)i;
  }
  __syncthreads();
  for (int k = 2; k <= N_; k <<= 1) {
    for (int j = k >> 1; j > 0; j >>= 1) {
      for (int i = tid; i < N_; i += 1024) {
        const int ij = i ^ j;
        if (ij > i) {
          const unsigned long long a = key[i], bb = key[ij];
          const bool up = ((i & k) == 0);
          if ((a > bb) == up) { key[i] = bb; key[ij] = a; }
        }
      }
      __syncthreads();
    }
  }
  for (int i = tid; i < S_; i += 1024)
    topidx[(long long)b * S_ + i] = (unsigned)(key[i] & 0xFFFFFFFFu);
}

// ---------------------------------------------------------------------------
// Kernel 4: gather selected tokens, recompute LN from saved stats, scale by
// softmax*N, add pos_embed, emit x_down (f32) and pos_down (i64).
// ---------------------------------------------------------------------------
__global__ __launch_bounds__(256) void gather_kernel(
    const float* __restrict__ x, const long long* __restrict__ pos,
    const float* __restrict__ pos_embed, const float* __restrict__ ln_w,
    const float* __restrict__ ln_b, const float* __restrict__ conf,
    const float* __restrict__ mu, const float* __restrict__ rstd,
    const float* __restrict__ bmax, const float* __restrict__ bsum,
    const unsigned* __restrict__ topidx, float* __restrict__ xout,
    long long* __restrict__ posout)
{
  const long long s = blockIdx.x;          // 0 .. B*S-1
  const int b = (int)(s / S_);
  const int c = threadIdx.x;
  const unsigned t = topidx[s];
  const long long tok = (long long)b * N_ + t;
  const float m = mu[tok], r = rstd[tok];
  const float soft = __expf(conf[tok] - bmax[b]) / bsum[b] * (float)N_;
  const long long p = pos[tok];
  const float xv = x[tok * C_ + c];
  const float xn = (xv - m) * r * ln_w[c] + ln_b[c];
  const float pe = pos_embed[p * C_ + c];
  xout[s * C_ + c] = xn * soft + pe;
  if (c == 0) posout[s] = p;
}

// ---------------------------------------------------------------------------
extern "C" void kernel_launch(void* const* d_in, const int* in_sizes, int n_in,
                              void* d_out, int out_size, void* d_ws, size_t ws_size,
                              hipStream_t stream)
{
  (void)in_sizes; (void)n_in; (void)out_size; (void)ws_size;
  const float*     x         = (const float*)d_in[0];
  const long long* pos       = (const long long*)d_in[1];
  const float*     pos_embed = (const float*)d_in[2];
  const float*     ln_w      = (const float*)d_in[3];
  const float*     ln_b      = (const float*)d_in[4];
  const float*     conf_w    = (const float*)d_in[5];
  const float*     conf_b    = (const float*)d_in[6];

  float* ws    = (float*)d_ws;
  float* conf  = ws;                                   // B*N
  float* mu    = conf + (size_t)B_ * N_;               // B*N
  float* rstd  = mu   + (size_t)B_ * N_;               // B*N
  float* bmax  = rstd + (size_t)B_ * N_;               // B
  float* bsum  = bmax + B_;                            // B
  unsigned* topidx = (unsigned*)(bsum + B_);           // B*S

  float*     xout   = (float*)d_out;                             // B*S*C f32
  long long* posout = (long long*)(xout + (size_t)B_ * S_ * C_); // B*S i64

  stats_kernel<<<dim3(B_ * N_ / (16 * TILES_)), dim3(32), 0, stream>>>(
      x, ln_w, ln_b, conf_w, conf_b, conf, mu, rstd);
  softmax_stats_kernel<<<dim3(B_), dim3(256), 0, stream>>>(conf, bmax, bsum);
  topk_sort_kernel<<<dim3(B_), dim3(1024), (size_t)N_ * 8, stream>>>(conf, topidx);
  gather_kernel<<<dim3(B_ * S_), dim3(256), 0, stream>>>(
      x, pos, pos_embed, ln_w, ln_b, conf, mu, rstd, bmax, bsum, topidx,
      xout, posout);
}